// RankingBCELoss_83253646066322
// MI455X (gfx1250) — compile-verified
//
#include <hip/hip_runtime.h>
#include <math.h>

typedef __attribute__((ext_vector_type(2))) float v2f;
typedef __attribute__((ext_vector_type(8))) float v8f;

#define N_ELEMS 16384
#define BIGF 1.0e6f
#define LOG2E 1.44269504089f
#define LN2   0.69314718056f

#define WAVES_PER_BLOCK 8
#define BLOCK_THREADS 256
#define MAIN_BLOCKS 2048
#define TOTAL_WAVES (MAIN_BLOCKS * WAVES_PER_BLOCK)   // 16384

#define SCAN_THREADS 1024
#define ELEMS_PER_THREAD (N_ELEMS / SCAN_THREADS)      // 16

// ---------------------------------------------------------------------------
// Pass 1: deterministic compaction (single block, LDS prefix scan, no atomics).
//   posA[0..P)   = x_i where t_i==1, padded to P16*16 with +BIG
//   negC[0..M)   = -x_j where t_j==0, padded to M16*16 with +BIG
// A (pos,neg) pair contributes softplus(-(x_i - x_j)) = softplus(-(posA+negC)).
// Any pair touching a +BIG pad has diff >= BIG -> softplus underflows to 0.
// ---------------------------------------------------------------------------
__global__ void __launch_bounds__(SCAN_THREADS)
compact_kernel(const float* __restrict__ x, const int* __restrict__ t,
               float* __restrict__ posA, float* __restrict__ negC,
               int* __restrict__ meta) {
    __shared__ int s[SCAN_THREADS];
    const int tid  = threadIdx.x;
    const int base = tid * ELEMS_PER_THREAD;

    int cnt = 0;
    #pragma unroll
    for (int k = 0; k < ELEMS_PER_THREAD; ++k) cnt += t[base + k];
    s[tid] = cnt;
    __syncthreads();

    // Hillis-Steele inclusive scan over 1024 thread-chunk counts
    for (int off = 1; off < SCAN_THREADS; off <<= 1) {
        int add = (tid >= off) ? s[tid - off] : 0;
        __syncthreads();
        s[tid] += add;
        __syncthreads();
    }

    const int P        = s[SCAN_THREADS - 1];
    const int posStart = s[tid] - cnt;          // positives before my chunk
    const int negStart = base - posStart;       // negatives before my chunk

    int p = posStart, n = negStart;
    #pragma unroll
    for (int k = 0; k < ELEMS_PER_THREAD; ++k) {
        float xv = x[base + k];
        if (t[base + k]) posA[p++] = xv;
        else             negC[n++] = -xv;
    }

    // pad tails to a multiple of 16 with sentinels; publish tile counts
    const int M   = N_ELEMS - P;
    const int P16 = (P + 15) >> 4;
    const int M16 = (M + 15) >> 4;
    if (tid < 16) {
        int pi = P + tid; if (pi < P16 * 16) posA[pi] = BIGF;
        int ni = M + tid; if (ni < M16 * 16) negC[ni] = BIGF;
    }
    if (tid == 0) { meta[0] = P16; meta[1] = M16; }
}

// ---------------------------------------------------------------------------
// Pass 2: dense P16 x M16 tile space. One V_WMMA_F32_16X16X4_F32 per 16x16
// tile builds  D[m][n] = posA_m * 1 + 1 * negC_n  (rank-2 product, C = 0),
// then a branch-free stable softplus on the 8 f32 accumulators per lane:
//   softplus(z) = max(z,0) + ln2 * log2(1 + exp2(-log2e*|z|))
// using raw v_exp_f32 / v_log_f32 (arg >= 1 -> no denorm guards needed).
// ---------------------------------------------------------------------------
__global__ void __launch_bounds__(BLOCK_THREADS)
pair_loss_kernel(const float* __restrict__ posA, const float* __restrict__ negC,
                 const int* __restrict__ meta, float* __restrict__ partials) {
    const int lane = threadIdx.x & 31;
    const int wave = threadIdx.x >> 5;
    const int P16 = meta[0];
    const int M16 = meta[1];
    const long long total = (long long)P16 * (long long)M16;
    const long long gid   = (long long)blockIdx.x * WAVES_PER_BLOCK + wave;

    float acc = 0.0f;
    if (total > 0 && gid < total) {
        // flat tile index -> (ti, tj); advance by stride with add/cond-subtract
        int ti = (int)(gid / M16);
        int tj = (int)(gid % M16);
        const int qs = TOTAL_WAVES / M16;
        const int rs = TOTAL_WAVES % M16;
        const int lsel = lane & 15;

        for (long long idx = gid; idx < total; idx += TOTAL_WAVES) {
            const float av = posA[ti * 16 + lsel];
            const float cv = negC[tj * 16 + lsel];

            v2f A;  // 32-bit 16x4 A layout: lanes 0-15 = K0/K1, lanes 16-31 = K2/K3 (zeroed)
            A.x = (lane < 16) ? av   : 0.0f;
            A.y = (lane < 16) ? 1.0f : 0.0f;
            v2f B;  // B rows: K0 = ones, K1 = negC
            B.x = (lane < 16) ? 1.0f : 0.0f;
            B.y = (lane < 16) ? cv   : 0.0f;

            v8f cz = {};
            v8f d = __builtin_amdgcn_wmma_f32_16x16x4_f32(
                false, A, false, B, (short)0, cz, false, false);

            // advance + speculative prefetch of next tile's fragments
            ti += qs; tj += rs;
            if (tj >= M16) { tj -= M16; ti += 1; }
            __builtin_prefetch(&negC[tj * 16 + lsel], 0, 0);
            __builtin_prefetch(&posA[ti * 16 + lsel], 0, 0);

            #pragma unroll
            for (int k = 0; k < 8; ++k) {
                float z = -d[k];
                float e = __builtin_amdgcn_exp2f(-LOG2E * __builtin_fabsf(z));
                float l = __builtin_amdgcn_logf(1.0f + e);   // log2, arg >= 1
                acc += fmaxf(z, 0.0f) + LN2 * l;
            }
        }
    }

    // wave32 shuffle tree, then fixed-order per-block partial (deterministic)
    for (int off = 16; off > 0; off >>= 1)
        acc += __shfl_down(acc, off, 32);

    __shared__ float lsum[WAVES_PER_BLOCK];
    if (lane == 0) lsum[wave] = acc;
    __syncthreads();
    if (threadIdx.x == 0) {
        float sblk = 0.0f;
        #pragma unroll
        for (int w = 0; w < WAVES_PER_BLOCK; ++w) sblk += lsum[w];
        partials[blockIdx.x] = sblk;
    }
}

// ---------------------------------------------------------------------------
// Pass 3: fixed-order reduction of partials + positive count -> final scalar.
// ---------------------------------------------------------------------------
__global__ void finalize_kernel(const float* __restrict__ partials,
                                const int* __restrict__ t,
                                float* __restrict__ out) {
    __shared__ float ssum[256];
    __shared__ int   scnt[256];
    float s = 0.0f;
    int cnt = 0;
    for (int i = threadIdx.x; i < MAIN_BLOCKS; i += 256) s += partials[i];
    for (int i = threadIdx.x; i < N_ELEMS; i += 256) cnt += t[i];
    ssum[threadIdx.x] = s;
    scnt[threadIdx.x] = cnt;
    __syncthreads();
    for (int st = 128; st > 0; st >>= 1) {
        if (threadIdx.x < st) {
            ssum[threadIdx.x] += ssum[threadIdx.x + st];
            scnt[threadIdx.x] += scnt[threadIdx.x + st];
        }
        __syncthreads();
    }
    if (threadIdx.x == 0) {
        long long P = (long long)scnt[0];
        long long npairs = P * (long long)(N_ELEMS - P);
        out[0] = (npairs > 0) ? (float)((double)ssum[0] / (double)npairs) : 0.0f;
    }
}

extern "C" void kernel_launch(void* const* d_in, const int* in_sizes, int n_in,
                              void* d_out, int out_size, void* d_ws, size_t ws_size,
                              hipStream_t stream) {
    (void)in_sizes; (void)n_in; (void)out_size; (void)ws_size;
    const float* x = (const float*)d_in[0];
    const int*   t = (const int*)d_in[1];
    float* out = (float*)d_out;

    float* posA     = (float*)d_ws;              // N floats (worst case P = N)
    float* negC     = posA + N_ELEMS;            // N floats
    float* partials = negC + N_ELEMS;            // MAIN_BLOCKS floats
    int*   meta     = (int*)(partials + MAIN_BLOCKS);  // 2 ints: P16, M16

    compact_kernel<<<1, SCAN_THREADS, 0, stream>>>(x, t, posA, negC, meta);
    pair_loss_kernel<<<MAIN_BLOCKS, BLOCK_THREADS, 0, stream>>>(posA, negC, meta, partials);
    finalize_kernel<<<1, 256, 0, stream>>>(partials, t, out);
}